// NEQUIPLayerFlax_40175124086945
// MI455X (gfx1250) — compile-verified
//
#include <hip/hip_runtime.h>
#include <math.h>

// NequIP layer for gfx1250: f32 WMMA (v_wmma_f32_16x16x4_f32) everywhere.
// Weights are pre-transposed to column-major in workspace so every WMMA
// B-operand is a single aligned 8-byte load.

#define NSPEC   5
#define CUTOFF  4.0f
#define SQRT3   1.7320508075688772f
#define PI_F    3.14159265358979323846f
#define HST     66   // LDS row stride (floats): bank-conflict-free + 8B aligned

typedef __attribute__((ext_vector_type(2))) float v2f;
typedef __attribute__((ext_vector_type(8))) float v8f;

static __device__ __forceinline__ v8f wmma4(v2f a, v2f b, v8f c) {
  // D = A(16x4) * B(4x16) + C(16x16), f32.
  return __builtin_amdgcn_wmma_f32_16x16x4_f32(false, a, false, b, (short)0, c,
                                               false, false);
}

static __device__ __forceinline__ float silu_f(float x) {
  return x / (1.0f + __expf(-x));
}

static __device__ __forceinline__ v2f ldb(const float* p) {
  return *(const v2f*)p;   // 8B-aligned by construction (k0 even, row*K even)
}

// ---------------------------------------------------------------------------
// Zero the node accumulator (N*512 floats) in workspace.
// ---------------------------------------------------------------------------
__global__ void zero_kernel(float4* __restrict__ p, int n4) {
  int i = blockIdx.x * blockDim.x + threadIdx.x;
  if (i < n4) p[i] = make_float4(0.f, 0.f, 0.f, 0.f);
}

// ---------------------------------------------------------------------------
// Batched weight transpose: src (B,K,N) -> dst (B,N,K).  batch = blockIdx.y.
// ---------------------------------------------------------------------------
__global__ void wtrans_kernel(const float* __restrict__ src,
                              float* __restrict__ dst, int K, int N) {
  const size_t boff = (size_t)blockIdx.y * K * N;
  const int i = blockIdx.x * blockDim.x + threadIdx.x;
  if (i < K * N) {
    const int k = i / N, n = i % N;
    dst[boff + (size_t)n * K + k] = src[boff + i];
  }
}

// ---------------------------------------------------------------------------
// Up-projection: s_up = s @ W_up_s * 1/8 ; v_up[:,:,x] = v[:,:,x] @ W_up_v / 8
// One wave = one 16-node tile. WMMA f32 16x16x4, K=64 (16 k-steps).
// su: (N,64)  vu: (N,64,3) row-major.  WT* are column-major (64,64).
// ---------------------------------------------------------------------------
__global__ __launch_bounds__(128) void up_kernel(
    const float* __restrict__ feats, const float* __restrict__ WTus,
    const float* __restrict__ WTuv, float* __restrict__ su,
    float* __restrict__ vu, int nNodes) {
  const int lane = threadIdx.x & 31;
  const int w    = threadIdx.x >> 5;
  const int base = (blockIdx.x * (blockDim.x >> 5) + w) * 16;
  if (base >= nNodes) return;
  const int row  = lane & 15;       // A row / B,D column-in-tile
  const int half = lane >> 4;
  const int kh   = half * 2;        // K sub-offset per A/B layout
  const float inv_c = 0.125f;       // 1/sqrt(64)
  const float* fR = feats + (size_t)(base + row) * 256;

  for (int t = 0; t < 4; ++t) {
    const int col = t * 16 + row;   // output channel d
    const float* bs = WTus + (size_t)col * 64;
    const float* bv = WTuv + (size_t)col * 64;
    v8f as = {0,0,0,0,0,0,0,0};
    v8f a0 = as, a1 = as, a2 = as;
    for (int ks = 0; ks < 16; ++ks) {
      const int k0 = ks * 4 + kh;
      v2f A;  A.x  = fR[k0];                A.y  = fR[k0 + 1];
      v2f A0; A0.x = fR[64 + 3*k0 + 0];     A0.y = fR[64 + 3*k0 + 3];
      v2f A1; A1.x = fR[64 + 3*k0 + 1];     A1.y = fR[64 + 3*k0 + 4];
      v2f A2; A2.x = fR[64 + 3*k0 + 2];     A2.y = fR[64 + 3*k0 + 5];
      const v2f Bs = ldb(bs + k0);
      const v2f Bv = ldb(bv + k0);
      as = wmma4(A,  Bs, as);
      a0 = wmma4(A0, Bv, a0);
      a1 = wmma4(A1, Bv, a1);
      a2 = wmma4(A2, Bv, a2);
    }
    const int rbase = base + half * 8;
#pragma unroll
    for (int r = 0; r < 8; ++r) {
      const int n = rbase + r;
      su[(size_t)n * 64  + col]       = as[r] * inv_c;
      vu[(size_t)n * 192 + col*3 + 0] = a0[r] * inv_c;
      vu[(size_t)n * 192 + col*3 + 1] = a1[r] * inv_c;
      vu[(size_t)n * 192 + col*3 + 2] = a2[r] * inv_c;
    }
  }
}

// ---------------------------------------------------------------------------
// Fused edge kernel: Bessel basis -> 3-layer MLP (WMMA) -> scale message
// components -> atomic scatter-add into node accumulator acc (N,512):
//   acc[:, 0:128]     = segment_sum(sc)
//   acc[:, 128+c*3+x] = segment_sum(vc[c,x]),  c in [0,128)
// One wave = 16 edges. LDS used for the layer transposes (stride-66 pad).
// WT0 (64,8), WT1 (64,64), WT2 (256,64) are column-major.
// ---------------------------------------------------------------------------
#define EW 4
__global__ __launch_bounds__(128) void edge_kernel(
    const float* __restrict__ vectors, const int* __restrict__ senders,
    const int* __restrict__ receivers, const float* __restrict__ WT0,
    const float* __restrict__ WT1, const float* __restrict__ WT2,
    const float* __restrict__ su, const float* __restrict__ vu,
    float* __restrict__ acc, int nEdges) {
  __shared__ float h1[EW][16 * HST];
  __shared__ float h2[EW][16 * HST];
  __shared__ float sYv[EW][16 * 3];
  __shared__ int   sSnd[EW][16];
  __shared__ int   sRcv[EW][16];

  const int lane  = threadIdx.x & 31;
  const int w     = threadIdx.x >> 5;
  const int ebase = (blockIdx.x * EW + w) * 16;
  if (ebase + 16 > nEdges) return;      // full tiles only (E % 64 == 0)
  const int row  = lane & 15;
  const int half = lane >> 4;
  const int kh   = half * 2;

  // --- per-edge geometry (each half-wave redundantly covers 16 edges) ---
  const int e = ebase + row;
  const float vx = vectors[3*e + 0];
  const float vy = vectors[3*e + 1];
  const float vz = vectors[3*e + 2];
  const float len  = sqrtf(vx*vx + vy*vy + vz*vz);
  const float invl = (len > 0.f) ? (1.0f / len) : 0.0f;
  if (lane < 16) {
    sYv[w][row*3 + 0] = SQRT3 * vx * invl;
    sYv[w][row*3 + 1] = SQRT3 * vy * invl;
    sYv[w][row*3 + 2] = SQRT3 * vz * invl;
    sSnd[w][row] = senders[e];
    sRcv[w][row] = receivers[e];
  }

  // --- radial basis (folded: 1/len, envelope, sqrt(2/cutoff), 1/sqrt(8)) ---
  const float u  = len * (1.0f / CUTOFF);
  const float u2 = u * u;
  const float u6 = u2 * u2 * u2;
  float env = 1.0f - 28.0f*u6 + 48.0f*u6*u - 21.0f*u6*u2;
  env = (u < 1.0f) ? env : 0.0f;
  const float pref = 0.70710678f * invl * env * 0.35355339f;

  const v8f vzero = {0,0,0,0,0,0,0,0};

  // --- layer 1: (16x8) @ (8x64) ---
  v8f d1[4];
#pragma unroll
  for (int t = 0; t < 4; ++t) d1[t] = vzero;
#pragma unroll
  for (int ks = 0; ks < 2; ++ks) {
    const int k0 = ks * 4 + kh;
    v2f A;
    A.x = pref * __sinf((float)(k0 + 1) * PI_F * u);
    A.y = pref * __sinf((float)(k0 + 2) * PI_F * u);
#pragma unroll
    for (int t = 0; t < 4; ++t)
      d1[t] = wmma4(A, ldb(WT0 + (t*16 + row)*8 + k0), d1[t]);
  }
#pragma unroll
  for (int t = 0; t < 4; ++t)
#pragma unroll
    for (int r = 0; r < 8; ++r)
      h1[w][(r + half*8)*HST + t*16 + row] = silu_f(d1[t][r]);
  __syncthreads();

  // --- layer 2: (16x64) @ (64x64), scale 1/8 inside silu ---
  v8f d2[4];
#pragma unroll
  for (int t = 0; t < 4; ++t) d2[t] = vzero;
  for (int ks = 0; ks < 16; ++ks) {
    const int k0 = ks * 4 + kh;
    v2f A; A.x = h1[w][row*HST + k0]; A.y = h1[w][row*HST + k0 + 1];
#pragma unroll
    for (int t = 0; t < 4; ++t)
      d2[t] = wmma4(A, ldb(WT1 + (t*16 + row)*64 + k0), d2[t]);
  }
#pragma unroll
  for (int t = 0; t < 4; ++t)
#pragma unroll
    for (int r = 0; r < 8; ++r)
      h2[w][(r + half*8)*HST + t*16 + row] = silu_f(d2[t][r] * 0.125f);
  __syncthreads();

  // --- layer 3: (16x64) @ (64x256) -> mix, consumed straight from D regs ---
  const float invs3 = 1.0f / SQRT3;
  for (int t = 0; t < 16; ++t) {
    v8f d3 = vzero;
    const float* bW = WT2 + (size_t)(t*16 + row) * 64;
    for (int ks = 0; ks < 16; ++ks) {
      const int k0 = ks * 4 + kh;
      v2f A; A.x = h2[w][row*HST + k0]; A.y = h2[w][row*HST + k0 + 1];
      d3 = wmma4(A, ldb(bW + k0), d3);
    }
    const int ch = t * 16 + row;   // mix channel; class uniform per tile t
#pragma unroll
    for (int r = 0; r < 8; ++r) {
      const int le  = half * 8 + r;           // local edge 0..15
      const float val = d3[r] * 0.125f;       // mix value (1/sqrt(64))
      const int snd = sSnd[w][le];
      const int rcv = sRcv[w][le];
      float* accR = acc + (size_t)rcv * 512;
      const float* suR = su + (size_t)snd * 64;
      const float* vuR = vu + (size_t)snd * 192;
      const float Y0 = sYv[w][le*3+0], Y1 = sYv[w][le*3+1], Y2 = sYv[w][le*3+2];
      if (t < 4) {                 // sc[0:64] = ms * mix
        atomicAdd(&accR[ch], suR[ch] * val);
      } else if (t < 8) {          // sc[64:128] = dot(mv,Y)/sqrt3 * mix
        const int c = ch - 64;
        const float dot = (vuR[c*3]*Y0 + vuR[c*3+1]*Y1 + vuR[c*3+2]*Y2) * invs3;
        atomicAdd(&accR[ch], dot * val);
      } else if (t < 12) {         // vc[0:64] = mv * mix
        const int c = ch - 128;
        atomicAdd(&accR[128 + c*3 + 0], vuR[c*3+0] * val);
        atomicAdd(&accR[128 + c*3 + 1], vuR[c*3+1] * val);
        atomicAdd(&accR[128 + c*3 + 2], vuR[c*3+2] * val);
      } else {                     // vc[64:128] = (ms outer Y) * mix
        const int c = ch - 192;
        const float msv = suR[c] * val;
        atomicAdd(&accR[128 + (64+c)*3 + 0], msv * Y0);
        atomicAdd(&accR[128 + (64+c)*3 + 1], msv * Y1);
        atomicAdd(&accR[128 + (64+c)*3 + 2], msv * Y2);
      }
    }
  }
}

// ---------------------------------------------------------------------------
// Down-projection + per-species skip (masked-A WMMA over 5 species) + gating.
// One wave = one 16-node tile. Scales folded into A so down-path and skip-path
// accumulate into the same D registers.  WTd* are (64,128) col-major; WTs* are
// (5,64,64) col-major per species.
// ---------------------------------------------------------------------------
__global__ __launch_bounds__(128) void down_kernel(
    const float* __restrict__ acc, const float* __restrict__ feats,
    const int* __restrict__ specie,
    const float* __restrict__ WTds, const float* __restrict__ WTdg,
    const float* __restrict__ WTdv,
    const float* __restrict__ WTss, const float* __restrict__ WTsg,
    const float* __restrict__ WTsv,
    float* __restrict__ out, int nNodes) {
  const int lane = threadIdx.x & 31;
  const int w    = threadIdx.x >> 5;
  const int base = (blockIdx.x * (blockDim.x >> 5) + w) * 16;
  if (base >= nNodes) return;
  const int row  = lane & 15;
  const int half = lane >> 4;
  const int kh   = half * 2;
  const int nrow = base + row;
  const int myspec = specie[nrow];
  const float dscale = 0.25f * 0.08838834764831845f;  // 1/sqrt(16)*1/sqrt(128)
  const float sscale = 0.125f;                        // 1/sqrt(64)
  const float* accR = acc   + (size_t)nrow * 512;
  const float* fR   = feats + (size_t)nrow * 256;
  const v8f vzero = {0,0,0,0,0,0,0,0};

  for (int t = 0; t < 4; ++t) {
    const int col = t * 16 + row;
    const float* bds = WTds + (size_t)col * 128;
    const float* bdg = WTdg + (size_t)col * 128;
    const float* bdv = WTdv + (size_t)col * 128;
    v8f df = vzero, dg = vzero, dv0 = vzero, dv1 = vzero, dv2 = vzero;

    // down GEMMs: K = 128 over the aggregated messages
    for (int ks = 0; ks < 32; ++ks) {
      const int k0 = ks * 4 + kh;
      v2f A;  A.x  = accR[k0]         * dscale; A.y  = accR[k0+1]       * dscale;
      v2f A0; A0.x = accR[128+3*k0+0] * dscale; A0.y = accR[128+3*k0+3] * dscale;
      v2f A1; A1.x = accR[128+3*k0+1] * dscale; A1.y = accR[128+3*k0+4] * dscale;
      v2f A2; A2.x = accR[128+3*k0+2] * dscale; A2.y = accR[128+3*k0+5] * dscale;
      df  = wmma4(A,  ldb(bds + k0), df);
      dg  = wmma4(A,  ldb(bdg + k0), dg);
      const v2f Bv = ldb(bdv + k0);
      dv0 = wmma4(A0, Bv, dv0);
      dv1 = wmma4(A1, Bv, dv1);
      dv2 = wmma4(A2, Bv, dv2);
    }

    // species skip GEMMs: K = 64, A rows masked per species, summed over sp
    for (int ks = 0; ks < 16; ++ks) {
      const int k0 = ks * 4 + kh;
      v2f A;  A.x  = fR[k0]        * sscale; A.y  = fR[k0+1]      * sscale;
      v2f A0; A0.x = fR[64+3*k0+0] * sscale; A0.y = fR[64+3*k0+3] * sscale;
      v2f A1; A1.x = fR[64+3*k0+1] * sscale; A1.y = fR[64+3*k0+4] * sscale;
      v2f A2; A2.x = fR[64+3*k0+2] * sscale; A2.y = fR[64+3*k0+5] * sscale;
#pragma unroll
      for (int sp = 0; sp < NSPEC; ++sp) {
        const float m = (myspec == sp) ? 1.0f : 0.0f;
        v2f Am  = {A.x  * m, A.y  * m};
        v2f Am0 = {A0.x * m, A0.y * m};
        v2f Am1 = {A1.x * m, A1.y * m};
        v2f Am2 = {A2.x * m, A2.y * m};
        const size_t so = (size_t)sp * 4096 + (size_t)col * 64 + k0;
        df  = wmma4(Am,  ldb(WTss + so), df);
        dg  = wmma4(Am,  ldb(WTsg + so), dg);
        const v2f Bv = ldb(WTsv + so);
        dv0 = wmma4(Am0, Bv, dv0);
        dv1 = wmma4(Am1, Bv, dv1);
        dv2 = wmma4(Am2, Bv, dv2);
      }
    }

    const int rbase = base + half * 8;
#pragma unroll
    for (int r = 0; r < 8; ++r) {
      const int n = rbase + r;
      float* orow = out + (size_t)n * 256;
      const float gsil = silu_f(dg[r]);
      orow[col]            = silu_f(df[r]);
      orow[64 + col*3 + 0] = gsil * dv0[r];
      orow[64 + col*3 + 1] = gsil * dv1[r];
      orow[64 + col*3 + 2] = gsil * dv2[r];
    }
  }
}

// ---------------------------------------------------------------------------
extern "C" void kernel_launch(void* const* d_in, const int* in_sizes, int n_in,
                              void* d_out, int out_size, void* d_ws,
                              size_t ws_size, hipStream_t stream) {
  const float* vectors    = (const float*)d_in[0];
  const float* node_feats = (const float*)d_in[1];
  const int*   node_spec  = (const int*)  d_in[2];
  const int*   senders    = (const int*)  d_in[3];
  const int*   receivers  = (const int*)  d_in[4];
  const float* W_up_s     = (const float*)d_in[5];
  const float* W_up_v     = (const float*)d_in[6];
  const float* W_mlp0     = (const float*)d_in[7];
  const float* W_mlp1     = (const float*)d_in[8];
  const float* W_mlp2     = (const float*)d_in[9];
  const float* W_down_s   = (const float*)d_in[10];
  const float* W_down_g   = (const float*)d_in[11];
  const float* W_down_v   = (const float*)d_in[12];
  const float* W_skip_s   = (const float*)d_in[13];
  const float* W_skip_g   = (const float*)d_in[14];
  const float* W_skip_v   = (const float*)d_in[15];

  const int N = in_sizes[2];   // node_specie element count
  const int E = in_sizes[3];   // senders element count
  float* out = (float*)d_out;

  // workspace layout: su | vu | acc | transposed weights
  float* su   = (float*)d_ws;             // N*64
  float* vu   = su  + (size_t)N * 64;     // N*192
  float* accb = vu  + (size_t)N * 192;    // N*512
  float* wt   = accb + (size_t)N * 512;
  float* wt_us = wt;            // 64*64
  float* wt_uv = wt_us + 4096;  // 64*64
  float* wt_0  = wt_uv + 4096;  // 64*8
  float* wt_1  = wt_0  + 512;   // 64*64
  float* wt_2  = wt_1  + 4096;  // 256*64
  float* wt_ds = wt_2  + 16384; // 64*128
  float* wt_dg = wt_ds + 8192;
  float* wt_dv = wt_dg + 8192;
  float* wt_ss = wt_dv + 8192;  // 5*64*64
  float* wt_sg = wt_ss + 20480;
  float* wt_sv = wt_sg + 20480;

  // 0) transpose all weights to column-major (one-time per launch, tiny)
  wtrans_kernel<<<dim3(16, 1),  256, 0, stream>>>(W_up_s,   wt_us, 64, 64);
  wtrans_kernel<<<dim3(16, 1),  256, 0, stream>>>(W_up_v,   wt_uv, 64, 64);
  wtrans_kernel<<<dim3(2, 1),   256, 0, stream>>>(W_mlp0,   wt_0,  8, 64);
  wtrans_kernel<<<dim3(16, 1),  256, 0, stream>>>(W_mlp1,   wt_1,  64, 64);
  wtrans_kernel<<<dim3(64, 1),  256, 0, stream>>>(W_mlp2,   wt_2,  64, 256);
  wtrans_kernel<<<dim3(32, 1),  256, 0, stream>>>(W_down_s, wt_ds, 128, 64);
  wtrans_kernel<<<dim3(32, 1),  256, 0, stream>>>(W_down_g, wt_dg, 128, 64);
  wtrans_kernel<<<dim3(32, 1),  256, 0, stream>>>(W_down_v, wt_dv, 128, 64);
  wtrans_kernel<<<dim3(16, 5),  256, 0, stream>>>(W_skip_s, wt_ss, 64, 64);
  wtrans_kernel<<<dim3(16, 5),  256, 0, stream>>>(W_skip_g, wt_sg, 64, 64);
  wtrans_kernel<<<dim3(16, 5),  256, 0, stream>>>(W_skip_v, wt_sv, 64, 64);

  // 1) zero accumulator
  {
    const int n4 = (N * 512) / 4;
    zero_kernel<<<(n4 + 255) / 256, 256, 0, stream>>>((float4*)accb, n4);
  }
  // 2) up-projection (4 waves/block, 64 nodes per block)
  up_kernel<<<(N + 63) / 64, 128, 0, stream>>>(node_feats, wt_us, wt_uv, su,
                                               vu, N);
  // 3) fused edge MLP + message scatter (4 waves/block, 64 edges per block)
  edge_kernel<<<(E + 63) / 64, 128, 0, stream>>>(vectors, senders, receivers,
                                                 wt_0, wt_1, wt_2, su, vu,
                                                 accb, E);
  // 4) down-projection + species skip + gating
  down_kernel<<<(N + 63) / 64, 128, 0, stream>>>(accb, node_feats, node_spec,
                                                 wt_ds, wt_dg, wt_dv,
                                                 wt_ss, wt_sg, wt_sv,
                                                 out, N);
}